// DeepLagrangianNetwork_35579509080615
// MI455X (gfx1250) — compile-verified
//
#include <hip/hip_runtime.h>

// DeepLagrangianNetwork fused kernel for MI455X (gfx1250, wave32).
//
// Pipeline per row (N = 2^21):
//   h  = leaky_relu(x[4] @ W1^T[4x64] + b1)        -> V_WMMA_F32_16X16X4_F32
//   Ld = softplus(h @ Wd^T + bd)  (2 values)        -> packed-f32 VALU dots
//   Lo = h @ Wo^T + bo            (1 value)
//   H  = L L^T + 1e-9 I  (2x2, stored row-major as 4 floats)
//
// Traffic: 16B in + 16B out per row = 64 MiB -> ~2.7us HBM floor, but the
// kernel is instruction-issue bound (~65k waves), so every VALU op counts:
//  * stage-1 GEMM on the matrix pipe (f32 WMMA, K=4), C as inline-0 SRC2
//  * bias via v_pk_add_f32 (op_sel broadcast), leaky as pk_mul + dual v_max
//  * stage-2 dots forced to v_pk_fma_f32 via __builtin_elementwise_fma,
//    weights stay uniform -> s_load'd SGPR-pair operands

typedef __attribute__((ext_vector_type(2))) float v2f;
typedef __attribute__((ext_vector_type(4))) float v4f;
typedef __attribute__((ext_vector_type(8))) float v8f;

#define LDS_STRIDE 66            // 64 h values + 2 pad dwords: conflict-free b64 reads
#define ROWS_PER_WAVE 32
#define WAVES_PER_BLOCK 8
#define THREADS_PER_BLOCK (WAVES_PER_BLOCK * 32)

__device__ __forceinline__ float softplus_f(float v) {
  float s = __logf(1.0f + __expf(v));
  return v > 20.0f ? v : s;
}

__launch_bounds__(THREADS_PER_BLOCK)
__global__ void dln_h_kernel(const float* __restrict__ x,
                             const float* __restrict__ W1,
                             const float* __restrict__ b1,
                             const float* __restrict__ Wd,
                             const float* __restrict__ bd,
                             const float* __restrict__ Wo,
                             const float* __restrict__ bo,
                             float* __restrict__ out) {
  __shared__ float lds[WAVES_PER_BLOCK * ROWS_PER_WAVE * LDS_STRIDE];

  const int tid   = threadIdx.x;
  const int lane  = tid & 31;
  const int wave  = tid >> 5;
  const int l15   = lane & 15;
  const int khalf = (lane >> 4) * 2;               // K base: 0 (lanes 0-15) / 2 (16-31)
  const unsigned rowBase =
      (blockIdx.x * WAVES_PER_BLOCK + wave) * ROWS_PER_WAVE;

  float* hbuf = &lds[wave * (ROWS_PER_WAVE * LDS_STRIDE)];

  // ---- Preload W1^T B-fragments (4x16 f32 per N-tile) + packed bias pairs ----
  // f32 4x16 B layout (dual of documented 16x4 A layout):
  //   VGPR0: lanes 0-15 -> K=0, lanes 16-31 -> K=2 ; VGPR1: K=1 / K=3.
  // W1 is [64,4] row-major, so W1^T[k][n] = W1[n*4 + k] -> contiguous float2.
  v2f bfrag[4];
  v2f bb[4];                                       // {b1[n], b1[n]} per N-tile
#pragma unroll
  for (int nt = 0; nt < 4; ++nt) {
    const int n = nt * 16 + l15;
    bfrag[nt] = *(const v2f*)(W1 + n * 4 + khalf);
    const float cb = b1[n];
    bb[nt].x = cb;
    bb[nt].y = cb;                                 // lowers to op_sel broadcast
  }

  const v2f kSlope = {0.01f, 0.01f};

  // ---- Stage 1: h = leaky_relu(q @ W1^T + b1), two 16-row tiles per wave ----
#pragma unroll
  for (int tile = 0; tile < 2; ++tile) {
    const unsigned row = rowBase + tile * 16 + l15;
    // A 16x4 f32 fragment: lane m holds (K=khalf, K=khalf+1) of row m.
    const v2f a = *(const v2f*)(x + row * 4 + khalf);
    const int mbase = tile * 16 + ((lane >> 4) << 3);
#pragma unroll
    for (int nt = 0; nt < 4; ++nt) {
      v8f c = {};                                  // lowers to inline-0 SRC2
      v8f d = __builtin_amdgcn_wmma_f32_16x16x4_f32(
          false, a, false, bfrag[nt], (short)0, c, false, false);
      const int n = nt * 16 + l15;
#pragma unroll
      for (int p = 0; p < 4; ++p) {
        // rows mbase+2p / mbase+2p+1, same column n -> same bias: packable.
        v2f t;
        t.x = d[2 * p];
        t.y = d[2 * p + 1];
        t = t + bb[nt];                            // v_pk_add_f32 (op_sel bcast)
        const v2f u = t * kSlope;                  // v_pk_mul_f32
        // leaky = max(t, 0.01t); scalar maxes pair into v_dual_max_num_f32
        hbuf[(mbase + 2 * p) * LDS_STRIDE + n]     = fmaxf(t.x, u.x);
        hbuf[(mbase + 2 * p + 1) * LDS_STRIDE + n] = fmaxf(t.y, u.y);
      }
    }
  }

  // Same-wave LDS RAW: DS ops are in-order per wave; fence compiler + counter.
  asm volatile("s_wait_dscnt 0" ::: "memory");

  // ---- Stage 2: one lane per row; 3 packed dots of length 64 from LDS ----
  const float* hrow = hbuf + lane * LDS_STRIDE;
  v2f a0 = {0.0f, 0.0f};
  v2f a1 = {0.0f, 0.0f};
  v2f a2 = {0.0f, 0.0f};
#pragma unroll
  for (int k = 0; k < 32; ++k) {
    const v2f hv = *(const v2f*)(hrow + 2 * k);    // conflict-free ds_load_b64
    // Uniform weight addresses -> s_load_b64; pk_fma takes SGPR-pair operand.
    const v2f w0 = *(const v2f*)(Wd + 2 * k);
    const v2f w1 = *(const v2f*)(Wd + 64 + 2 * k);
    const v2f w2 = *(const v2f*)(Wo + 2 * k);
    a0 = __builtin_elementwise_fma(hv, w0, a0);    // v_pk_fma_f32 (forced)
    a1 = __builtin_elementwise_fma(hv, w1, a1);
    a2 = __builtin_elementwise_fma(hv, w2, a2);
  }
  const float acc0 = a0.x + a0.y;
  const float acc1 = a1.x + a1.y;
  const float acc2 = a2.x + a2.y;

  const float ld0 = softplus_f(acc0 + bd[0]);
  const float ld1 = softplus_f(acc1 + bd[1]);
  const float lo  = acc2 + bo[0];

  // H = L L^T + 1e-9 I with L = [[ld0, 0], [lo, ld1]]
  v4f H;
  H.x = fmaf(ld0, ld0, 1e-9f);
  H.y = ld0 * lo;
  H.z = ld0 * lo;
  H.w = fmaf(lo, lo, fmaf(ld1, ld1, 1e-9f));

  const unsigned row = rowBase + lane;
  *(v4f*)(out + row * 4) = H;                      // coalesced b128 store
}

extern "C" void kernel_launch(void* const* d_in, const int* in_sizes, int n_in,
                              void* d_out, int out_size, void* d_ws, size_t ws_size,
                              hipStream_t stream) {
  const float* x  = (const float*)d_in[0];
  const float* W1 = (const float*)d_in[1];
  const float* b1 = (const float*)d_in[2];
  const float* Wd = (const float*)d_in[3];
  const float* bd = (const float*)d_in[4];
  const float* Wo = (const float*)d_in[5];
  const float* bo = (const float*)d_in[6];
  float* out = (float*)d_out;

  const int nrows  = in_sizes[0] / 4;                            // 2,097,152
  const int blocks = nrows / (WAVES_PER_BLOCK * ROWS_PER_WAVE);  // 8192, exact
  dln_h_kernel<<<blocks, THREADS_PER_BLOCK, 0, stream>>>(
      x, W1, b1, Wd, bd, Wo, bo, out);
}